// StereoLoss_45406394253313
// MI455X (gfx1250) — compile-verified
//
#include <hip/hip_runtime.h>
#include <math.h>

// ---------------------------------------------------------------------------
// StereoLoss pipeline for MI455X (gfx1250, wave32).
//   K1: down2 (2x2 mean) of left/right  + zero reduction accumulators
//   K2: disparity warp (bilinear grid sample) of right -> recon, outlier sum
//   K3: 9x9 box sums via V_WMMA_F32_16X16X4_F32 banded-constant GEMMs,
//       LDS staging by Tensor Data Mover (tensor_load_to_lds) on interior tiles
//   K4: per-pixel LCN loss  |(lcn_l - lcn_r) * std_l|
//   K5: 12x12 adaptive-support-weight aggregation + per-batch reduction (VALU)
//   K6: finalize out[n] = asw_mean[n] + outlier_mean
// All intermediates live in d_ws (~15 MB -> L2 resident, 192 MB L2).
// ---------------------------------------------------------------------------

typedef float v2f __attribute__((ext_vector_type(2)));
typedef float v8f __attribute__((ext_vector_type(8)));
typedef unsigned u32x4 __attribute__((ext_vector_type(4)));
typedef unsigned u32x8 __attribute__((ext_vector_type(8)));

#define H1 768
#define W1 1216
#define H2 384
#define W2 608
#define HW2 (H2 * W2)          // 233472
#define NBATCH 2
#define MAXD 1216.0f

// workspace layout (float offsets)
#define OFS_LEFT   0
#define OFS_RIGHT  (OFS_LEFT  + NBATCH * HW2)
#define OFS_RECON  (OFS_RIGHT + NBATCH * HW2)
#define OFS_S1L    (OFS_RECON + NBATCH * HW2)
#define OFS_S2L    (OFS_S1L   + NBATCH * HW2)
#define OFS_S1R    (OFS_S2L   + NBATCH * HW2)
#define OFS_S2R    (OFS_S1R   + NBATCH * HW2)
#define OFS_LOSS   (OFS_S2R   + NBATCH * HW2)
#define OFS_ACC    (OFS_LOSS  + NBATCH * HW2)   // [0]=outlier sum, [1..2]=asw sums

// ------------------------------- K1: down2 ---------------------------------
__global__ void k_down2(const float* __restrict__ L, const float* __restrict__ R,
                        float* __restrict__ ws) {
  int t = blockIdx.x * blockDim.x + threadIdx.x;
  if (t < 4) ws[OFS_ACC + t] = 0.0f;            // deterministic re-init per launch
  if (t >= 4 * HW2) return;
  int plane = t / HW2;                          // 0,1: left n=0,1 ; 2,3: right
  int pix = t - plane * HW2;
  int y = pix / W2, x = pix - (pix / W2) * W2;
  const float* src = (plane < 2 ? L : R) + (plane & 1) * (H1 * W1);
  const float* p0 = src + (2 * y) * W1 + 2 * x;
  float v = 0.25f * (p0[0] + p0[1] + p0[W1] + p0[W1 + 1]);
  float* dst = ws + (plane < 2 ? OFS_LEFT : OFS_RIGHT) + (plane & 1) * HW2;
  dst[pix] = v;
}

// --------------------- K2: grid-sample warp + outlier ----------------------
__global__ void k_recon(const float* __restrict__ xpos, float* __restrict__ ws) {
  int t = blockIdx.x * blockDim.x + threadIdx.x;
  float outl = 0.0f;
  if (t < NBATCH * HW2) {
    int n = t / HW2, pix = t - n * HW2;
    int y = pix / W2, x = pix - (pix / W2) * W2;
    float xp = xpos[t];
    float disp = -(xp * (float)W2 - (float)x);
    float gx = (-1.0f + 2.0f * (float)x / (float)(W2 - 1)) - disp * (2.0f / (float)W2);
    float gy = -1.0f + 2.0f * (float)y / (float)(H2 - 1);
    float ix = (gx + 1.0f) * 0.5f * (float)(W2 - 1);
    float iy = (gy + 1.0f) * 0.5f * (float)(H2 - 1);
    float fx0 = floorf(ix), fy0 = floorf(iy);
    int x0 = (int)fx0, y0 = (int)fy0;
    float wx1 = ix - fx0, wx0 = 1.0f - wx1;
    float wy1 = iy - fy0, wy0 = 1.0f - wy1;
    const float* rp = ws + OFS_RIGHT + n * HW2;
    auto samp = [&](int yy, int xx) -> float {
      bool ok = (xx >= 0) & (xx < W2) & (yy >= 0) & (yy < H2);
      int yc = min(max(yy, 0), H2 - 1), xc = min(max(xx, 0), W2 - 1);
      float v = rp[yc * W2 + xc];
      return ok ? v : 0.0f;
    };
    float rec = samp(y0, x0) * wy0 * wx0 + samp(y0, x0 + 1) * wy0 * wx1 +
                samp(y0 + 1, x0) * wy1 * wx0 + samp(y0 + 1, x0 + 1) * wy1 * wx1;
    ws[OFS_RECON + t] = rec;
    outl = disp < 0.0f ? -disp : (disp > MAXD ? disp - MAXD : 0.0f);
  }
  // wave32 reduction, one atomic per wave
  #pragma unroll
  for (int o = 16; o > 0; o >>= 1) outl += __shfl_down(outl, o, 32);
  if ((threadIdx.x & 31) == 0) atomicAdd(ws + OFS_ACC + 0, outl);
}

// ------------- K3: 9x9 box sums via WMMA banded-matrix GEMMs ---------------
// One wave computes one 16x16 output tile of one plane.
//   T(16x24)  = Wv(16x24) * Iext(24x24)   (vertical 9-tap sums)
//   Out(16x16)= T(16x24)  * Wh(24x16)     (horizontal 9-tap sums)
// Band matrices Wv/Wh (Wx[m][k]=1 iff m<=k<=m+8) are built in registers.
// A 16x4 f32 fragment layout: lane -> M = lane&15, VGPR j -> K = 4*kc + j + 2*(lane>>4).
// B 4x16 mirrors it with lane -> N.
// Interior tiles stage the 24x24 halo'd input via the Tensor Data Mover
// (tensor_load_to_lds, TENSORcnt); border tiles use a bounds-checked fill.
#define SISTR 24
__global__ void k_box81(float* __restrict__ ws) {
  __shared__ float sI[8][24 * SISTR + 16];  // +16 slack for dead N=24..31 frag cols
  __shared__ float sT[8][16 * 32];
  const int lane = threadIdx.x & 31;
  const int wv = threadIdx.x >> 5;
  const int p = blockIdx.y;                     // plane 0..7
  const int img = p >> 2, sq = (p >> 1) & 1, n = p & 1;
  const float* in = ws + (img ? OFS_RECON : OFS_LEFT) + n * HW2;
  float* out = ws + (img == 0 ? (sq == 0 ? OFS_S1L : OFS_S2L)
                              : (sq == 0 ? OFS_S1R : OFS_S2R)) + n * HW2;
  const int tt = blockIdx.x * 8 + wv;           // 0..911 (24x38 tiles)
  const int tr = tt / (W2 / 16), tc = tt - tr * (W2 / 16);
  const int r0 = tr * 16, c0 = tc * 16;

  const bool interior = (tr > 0) & (tr < H2 / 16 - 1) & (tc > 0) & (tc < W2 / 16 - 1);
  if (interior) {
    // --- TDM: DMA 24 rows x 24 cols (stride W2) from global into LDS ------
    unsigned long long ga =
        (unsigned long long)(uintptr_t)(in + (r0 - 4) * W2 + (c0 - 4));
    unsigned lds_off = (unsigned)(uintptr_t)&sI[wv][0];  // flat LDS low 32 = offset
    u32x4 g0;
    g0[0] = 1u;                                   // count=1, flags 0
    g0[1] = lds_off;                              // lds_addr (bytes)
    g0[2] = (unsigned)(ga & 0xffffffffu);         // global_addr[31:0]
    g0[3] = (unsigned)((ga >> 32) & 0x01ffffffu)  // global_addr[56:32]
          | (2u << 30);                           // type=2 ("image")
    u32x8 g1;
    g1[0] = (2u << 16);        // workgroup_mask=0, data_size=2 (4 bytes)
    g1[1] = (24u << 16);       // atomic_barrier_addr=0 | tensor_dim0[15:0]=24
    g1[2] = (24u << 16);       // tensor_dim0[31:16]=0 | tensor_dim1[15:0]=24
    g1[3] = (24u << 16);       // tensor_dim1[31:16]=0 | tile_dim0=24
    g1[4] = 24u;               // tile_dim1=24, tile_dim2=0
    g1[5] = (unsigned)W2;      // tensor_dim0_stride[31:0] (data_size units)
    g1[6] = 0u;                // stride hi, tensor_dim1_stride lo (unused, 2D)
    g1[7] = 0u;
    asm volatile("tensor_load_to_lds %0, %1" :: "s"(g0), "s"(g1) : "memory");
    __builtin_amdgcn_s_wait_tensorcnt(0);
  } else {
    // --- border tiles: bounds-checked fill with zero padding --------------
    for (int i = lane; i < 24 * SISTR; i += 32) {
      int row = i / SISTR, col = i - row * SISTR;
      int gr = r0 - 4 + row, gc = c0 - 4 + col;
      float v = 0.0f;
      if (gr >= 0 && gr < H2 && gc >= 0 && gc < W2) v = in[gr * W2 + gc];
      sI[wv][i] = v;
    }
  }

  const int MN = lane & 15;
  const int hi = lane >> 4;

  // vertical pass: T = Wv * Iext  (two 16-wide N blocks, K=24 -> 6 WMMAs each)
  for (int nb = 0; nb < 2; ++nb) {
    v8f acc = {};
    #pragma unroll
    for (int kc = 0; kc < 6; ++kc) {
      v2f a, b;
      #pragma unroll
      for (int j = 0; j < 2; ++j) {
        int k = kc * 4 + j + 2 * hi;
        a[j] = (k >= MN && k <= MN + 8) ? 1.0f : 0.0f;   // Wv band
        float bv = sI[wv][k * SISTR + nb * 16 + MN];     // Iext rows (raw)
        if (sq) bv *= bv;                                // square at frag read
        b[j] = bv;
      }
      acc = __builtin_amdgcn_wmma_f32_16x16x4_f32(false, a, false, b,
                                                  (short)0, acc, false, false);
    }
    #pragma unroll
    for (int r = 0; r < 8; ++r)
      sT[wv][(r + 8 * hi) * 32 + nb * 16 + MN] = acc[r];  // D layout -> LDS
  }

  // horizontal pass: Out = T * Wh  (K=24 -> 6 WMMAs)
  v8f acc = {};
  #pragma unroll
  for (int kc = 0; kc < 6; ++kc) {
    v2f a, b;
    #pragma unroll
    for (int j = 0; j < 2; ++j) {
      int k = kc * 4 + j + 2 * hi;
      a[j] = sT[wv][MN * 32 + k];                        // T rows (A layout)
      b[j] = (k >= MN && k <= MN + 8) ? 1.0f : 0.0f;     // Wh band
    }
    acc = __builtin_amdgcn_wmma_f32_16x16x4_f32(false, a, false, b,
                                                (short)0, acc, false, false);
  }
  #pragma unroll
  for (int r = 0; r < 8; ++r)
    out[(r0 + r + 8 * hi) * W2 + (c0 + MN)] = acc[r];    // box SUM (÷81 later)
}

// ----------------------- K4: LCN loss per pixel ----------------------------
__global__ void k_losses(float* __restrict__ ws) {
  int t = blockIdx.x * blockDim.x + threadIdx.x;
  if (t >= NBATCH * HW2) return;
  const float inv81 = 1.0f / 81.0f;
  float l = ws[OFS_LEFT + t];
  float r = ws[OFS_RECON + t];
  float m1 = ws[OFS_S1L + t] * inv81;
  float q1 = ws[OFS_S2L + t] * inv81;
  float m2 = ws[OFS_S1R + t] * inv81;
  float q2 = ws[OFS_S2R + t] * inv81;
  float sd1 = sqrtf(fmaxf(q1 - m1 * m1, 1e-6f));
  float sd2 = sqrtf(fmaxf(q2 - m2 * m2, 1e-6f));
  float lcnL = (l - m1) / (sd1 + 1e-6f);
  float lcnR = (r - m2) / (sd2 + 1e-6f);
  ws[OFS_LOSS + t] = fabsf((lcnL - lcnR) * sd1);
}

// ---------------- K5: adaptive support weights + reduction -----------------
#define KW 12
#define PADT 6          // pad = (6,5) in both dims -> patch rows y-6..y+5
#define TILE 16
#define EXT (TILE + KW - 1)   // 27
#define ESTR 28

__global__ void k_asw(float* __restrict__ ws) {
  __shared__ float sImg[EXT * ESTR];
  __shared__ float sCst[EXT * ESTR];
  __shared__ float sRed[8];
  const int n = blockIdx.z;
  const int r0 = blockIdx.y * TILE, c0 = blockIdx.x * TILE;
  const float* img = ws + OFS_LEFT + n * HW2;
  const float* cst = ws + OFS_LOSS + n * HW2;
  for (int i = threadIdx.x; i < EXT * ESTR; i += blockDim.x) {
    int row = i / ESTR, col = i - (i / ESTR) * ESTR;
    int gr = r0 - PADT + row, gc = c0 - PADT + col;
    bool ok = (col < EXT) && gr >= 0 && gr < H2 && gc >= 0 && gc < W2;
    float vi = 0.0f, vc = 0.0f;
    if (ok) { int g = gr * W2 + gc; vi = img[g]; vc = cst[g]; }
    sImg[i] = vi; sCst[i] = vc;
  }
  __builtin_prefetch(img + (r0 + TILE) * W2 + c0, 0, 0);  // global_prefetch_b8
  __syncthreads();
  int ty = threadIdx.x >> 4, tx = threadIdx.x & 15;       // 256 thr = 16x16 tile
  float center = sImg[(ty + PADT) * ESTR + (tx + PADT)];
  float num = 0.0f, den = 0.0f;
  #pragma unroll 4
  for (int dy = 0; dy < KW; ++dy) {
    int base = (ty + dy) * ESTR + tx;
    #pragma unroll
    for (int dx = 0; dx < KW; ++dx) {
      float pv = sImg[base + dx];
      float cv = sCst[base + dx];
      float w = __expf(-fabsf(center - pv) * 0.5f);
      num = fmaf(w, cv, num);
      den += w;
    }
  }
  float val = num / den;
  #pragma unroll
  for (int o = 16; o > 0; o >>= 1) val += __shfl_down(val, o, 32);
  if ((threadIdx.x & 31) == 0) sRed[threadIdx.x >> 5] = val;
  __syncthreads();
  if (threadIdx.x == 0) {
    float s = 0.0f;
    #pragma unroll
    for (int i = 0; i < 8; ++i) s += sRed[i];
    atomicAdd(ws + OFS_ACC + 1 + n, s);
  }
}

// ------------------------------ K6: finalize -------------------------------
__global__ void k_final(const float* __restrict__ ws, float* __restrict__ out) {
  int n = threadIdx.x;
  if (n < NBATCH) {
    float outlier_mean = ws[OFS_ACC + 0] / (float)(NBATCH * HW2);
    out[n] = ws[OFS_ACC + 1 + n] / (float)HW2 + outlier_mean;
  }
}

// ---------------------------------------------------------------------------
extern "C" void kernel_launch(void* const* d_in, const int* in_sizes, int n_in,
                              void* d_out, int out_size, void* d_ws, size_t ws_size,
                              hipStream_t stream) {
  const float* L  = (const float*)d_in[0];
  const float* R  = (const float*)d_in[1];
  const float* XP = (const float*)d_in[2];
  float* ws  = (float*)d_ws;
  float* out = (float*)d_out;

  k_down2 <<<(4 * HW2 + 255) / 256, 256, 0, stream>>>(L, R, ws);
  k_recon <<<(NBATCH * HW2 + 255) / 256, 256, 0, stream>>>(XP, ws);
  dim3 g3((H2 / 16) * (W2 / 16) / 8, 8, 1);   // 114 blocks x 8 planes, 8 waves/blk
  k_box81 <<<g3, 256, 0, stream>>>(ws);
  k_losses<<<(NBATCH * HW2 + 255) / 256, 256, 0, stream>>>(ws);
  dim3 g5(W2 / 16, H2 / 16, NBATCH);
  k_asw   <<<g5, 256, 0, stream>>>(ws);
  k_final <<<1, 32, 0, stream>>>(ws, out);
}